// PairFeatureNet_12618613915748
// MI455X (gfx1250) — compile-verified
//
#include <hip/hip_runtime.h>
#include <math.h>

typedef float v2f __attribute__((ext_vector_type(2)));
typedef float v4f __attribute__((ext_vector_type(4)));
typedef float v8f __attribute__((ext_vector_type(8)));

#define N_RES 768
#define CS    384
#define CP    128
#define KREL  32
#define NBIN  65
#define EPSV  1e-10f

// ---------------------------------------------------------------------------
// Kernel 1: P = s @ W^T + b  via V_WMMA_F32_16X16X4_F32.
// One wave (32 threads) computes one 16x16 tile of P, looping K in steps of 4.
// blockIdx.z selects the i-projection (W_i/b_i -> Pi) or j-projection.
//
// A tile (16x4 f32, ISA 7.12.2): lane = M (lane%16); VGPR0 holds K=0 (lanes
// 0-15) / K=2 (lanes 16-31), VGPR1 holds K=1 / K=3.
// B tile (4x16 f32): N = lane%16; VGPR0 holds K=0 / K=2, VGPR1 K=1 / K=3.
// B[k][n] = W[n][k]  (since P = s @ W^T).
// D tile (16x16 f32): VGPR r, lanes 0-15 -> M=r, lanes 16-31 -> M=r+8, N=lane%16.
// ---------------------------------------------------------------------------
__global__ __launch_bounds__(32) void proj_gemm_wmma(
    const float* __restrict__ s,
    const float* __restrict__ W_i, const float* __restrict__ b_i,
    const float* __restrict__ W_j, const float* __restrict__ b_j,
    float* __restrict__ Pi, float* __restrict__ Pj)
{
    const float* W    = (blockIdx.z == 0) ? W_i : W_j;
    const float* bias = (blockIdx.z == 0) ? b_i : b_j;
    float*       P    = (blockIdx.z == 0) ? Pi  : Pj;

    const int n0   = blockIdx.x * 16;   // output-channel tile
    const int row0 = blockIdx.y * 16;   // residue tile
    const int lane = threadIdx.x;       // 0..31, EXEC all-ones (full wave, no divergence)
    const int m    = lane & 15;
    const int half = lane >> 4;

    const v2f* srow = (const v2f*)(s + (size_t)(row0 + m) * CS);
    const v2f* wrow = (const v2f*)(W + (size_t)(n0  + m) * CS);

    v8f acc = {};
    #pragma unroll 4
    for (int k2 = 0; k2 < CS / 2; k2 += 2) {      // k2 = k/2, K advances by 4
        v2f a = srow[k2 + half];                  // s[row0+m][k + 2*half .. +1]
        v2f b = wrow[k2 + half];                  // W[n0+m][k + 2*half .. +1]
        acc = __builtin_amdgcn_wmma_f32_16x16x4_f32(
            /*neg_a=*/false, a, /*neg_b=*/false, b,
            /*c_mod=*/(short)0, acc, /*reuse_a=*/false, /*reuse_b=*/false);
    }

    const float bv = bias[n0 + m];                // bias is per output channel N
    #pragma unroll
    for (int r = 0; r < 8; ++r) {
        const int row = row0 + r + 8 * half;
        P[(size_t)row * CP + (n0 + m)] = acc[r] + bv;
    }
}

// ---------------------------------------------------------------------------
// Kernel 2: Rel[bin][c] = W_rel[c][bin] + b_rel[c] + b_t[c]   (65 x 128)
// Folds both remaining biases so the pair kernel is pure FMA + store.
// ---------------------------------------------------------------------------
__global__ void prep_rel(const float* __restrict__ W_rel,
                         const float* __restrict__ b_rel,
                         const float* __restrict__ b_t,
                         float* __restrict__ Rel)
{
    const int idx = blockIdx.x * blockDim.x + threadIdx.x;
    if (idx < NBIN * CP) {
        const int bin = idx / CP;
        const int c   = idx % CP;
        Rel[idx] = W_rel[c * NBIN + bin] + b_rel[c] + b_t[c];
    }
}

// ---------------------------------------------------------------------------
// Kernel 3: the bandwidth kernel (302 MB of output -> ~13 us at 23.3 TB/s).
// Block = 256 threads: 8 j-values x 32 float4 channel-lanes. i is fixed per
// block, so p_i / W_t / trans[i] are loaded once into registers. Pj + Rel rows
// stay hot in L2 (<1 MB working set vs 192 MB L2). Output uses nontemporal
// b128 stores so the write-once stream doesn't evict the hot tables.
// ---------------------------------------------------------------------------
__global__ __launch_bounds__(256) void pair_fuse(
    const v4f* __restrict__ Pi, const v4f* __restrict__ Pj,
    const v4f* __restrict__ Rel, const v4f* __restrict__ Wt4,
    const float* __restrict__ trans, const float* __restrict__ mask,
    v4f* __restrict__ out)
{
    const int i  = blockIdx.y;
    const int jl = threadIdx.x >> 5;              // 0..7
    const int j  = blockIdx.x * 8 + jl;
    const int c4 = threadIdx.x & 31;              // float4 channel index 0..31

    const v4f pi = Pi[i * 32 + c4];
    const v4f wt = Wt4[c4];
    const float tix = trans[i * 3 + 0];
    const float tiy = trans[i * 3 + 1];
    const float tiz = trans[i * 3 + 2];

    const v4f pj = Pj[j * 32 + c4];
    int d = i - j;
    d = d < -KREL ? -KREL : (d > KREL ? KREL : d);
    const v4f rel = Rel[(d + KREL) * 32 + c4];

    const float dx = tix - trans[j * 3 + 0];
    const float dy = tiy - trans[j * 3 + 1];
    const float dz = tiz - trans[j * 3 + 2];
    const float dist = sqrtf(EPSV + dx * dx + dy * dy + dz * dz);
    const float msk  = mask[(size_t)i * N_RES + j];

    v4f o;
    o.x = (pi.x + pj.x + rel.x + dist * wt.x) * msk;
    o.y = (pi.y + pj.y + rel.y + dist * wt.y) * msk;
    o.z = (pi.z + pj.z + rel.z + dist * wt.z) * msk;
    o.w = (pi.w + pj.w + rel.w + dist * wt.w) * msk;

    const size_t oidx = ((size_t)i * N_RES + j) * 32 + c4;
    __builtin_nontemporal_store(o, &out[oidx]);
}

// ---------------------------------------------------------------------------
// Host launch. Input order (setup_inputs dict):
// 0:s 1:trans 2:p_mask 3:W_i 4:b_i 5:W_j 6:b_j 7:W_rel 8:b_rel 9:W_t 10:b_t
// ---------------------------------------------------------------------------
extern "C" void kernel_launch(void* const* d_in, const int* in_sizes, int n_in,
                              void* d_out, int out_size, void* d_ws, size_t ws_size,
                              hipStream_t stream) {
    const float* s     = (const float*)d_in[0];
    const float* trans = (const float*)d_in[1];
    const float* mask  = (const float*)d_in[2];
    const float* W_i   = (const float*)d_in[3];
    const float* b_i   = (const float*)d_in[4];
    const float* W_j   = (const float*)d_in[5];
    const float* b_j   = (const float*)d_in[6];
    const float* W_rel = (const float*)d_in[7];
    const float* b_rel = (const float*)d_in[8];
    const float* W_t   = (const float*)d_in[9];
    const float* b_t   = (const float*)d_in[10];
    float* out = (float*)d_out;

    // Workspace layout (bytes): Pi [0, 384K), Pj [384K, 768K), Rel [768K, +33K)
    char* ws = (char*)d_ws;
    float* Pi  = (float*)(ws);
    float* Pj  = (float*)(ws + (size_t)N_RES * CP * sizeof(float));
    float* Rel = (float*)(ws + (size_t)2 * N_RES * CP * sizeof(float));

    // 1) Projections via WMMA: grid (CP/16 n-tiles, N/16 m-tiles, 2 matrices)
    proj_gemm_wmma<<<dim3(CP / 16, N_RES / 16, 2), 32, 0, stream>>>(
        s, W_i, b_i, W_j, b_j, Pi, Pj);

    // 2) Relpos table with folded biases
    prep_rel<<<(NBIN * CP + 255) / 256, 256, 0, stream>>>(W_rel, b_rel, b_t, Rel);

    // 3) Fused broadcast + distance + mask, nontemporal b128 stores
    pair_fuse<<<dim3(N_RES / 8, N_RES), 256, 0, stream>>>(
        (const v4f*)Pi, (const v4f*)Pj, (const v4f*)Rel,
        (const v4f*)W_t, trans, mask, (v4f*)out);

    (void)in_sizes; (void)n_in; (void)out_size; (void)ws_size;
}